// ConditionalContrastiveLoss_1683627180664
// MI455X (gfx1250) — compile-verified
//
#include <hip/hip_runtime.h>
#include <hip/hip_bf16.h>
#include <math.h>
#include <stdint.h>

typedef __attribute__((ext_vector_type(2))) float v2f;
typedef __attribute__((ext_vector_type(8))) float v8f;

#define NROWS 8192
#define DDIM  128
#define INV_T 2.0f      // 1 / TEMPERATURE
#define EPSN  1e-8f

#define BLK_ROWS   128   // 8 waves * 16 rows
#define CHUNK_COLS 64
#define NCHUNKS    8     // cols per block = 512
#define KHALF      64    // K staged in two halves to bound LDS
#define ASTRIDE    68    // LDS row stride (floats): 4*row bank rotation, conflict-free

// ---------------------------------------------------------------------------
// Kernel A: per-row L2 normalize embed -> en; e2p[i] = exp(2*cos(en_i, pn_i));
// also zeros row_sum / pos_sum (graph-replay safe re-init).
// One wave32 per row; lane holds one float4 (32*4 = 128 = D).
// ---------------------------------------------------------------------------
__global__ __launch_bounds__(256) void normalize_kernel(
    const float* __restrict__ embed, const float* __restrict__ proxy,
    float* __restrict__ en, float* __restrict__ e2p,
    float* __restrict__ row_sum, float* __restrict__ pos_sum) {
  const int wave = threadIdx.x >> 5;
  const int lane = threadIdx.x & 31;
  const int row  = blockIdx.x * 8 + wave;
  const float4 e = reinterpret_cast<const float4*>(embed + (size_t)row * DDIM)[lane];
  const float4 p = reinterpret_cast<const float4*>(proxy + (size_t)row * DDIM)[lane];
  float se = e.x*e.x + e.y*e.y + e.z*e.z + e.w*e.w;
  float sp = p.x*p.x + p.y*p.y + p.z*p.z + p.w*p.w;
  float dp = e.x*p.x + e.y*p.y + e.z*p.z + e.w*p.w;
#pragma unroll
  for (int m = 16; m >= 1; m >>= 1) {
    se += __shfl_xor(se, m, 32);
    sp += __shfl_xor(sp, m, 32);
    dp += __shfl_xor(dp, m, 32);
  }
  const float inv_e = 1.0f / fmaxf(sqrtf(se), EPSN);
  float4 o;
  o.x = e.x * inv_e; o.y = e.y * inv_e; o.z = e.z * inv_e; o.w = e.w * inv_e;
  reinterpret_cast<float4*>(en + (size_t)row * DDIM)[lane] = o;
  if (lane == 0) {
    const float inv_p = 1.0f / fmaxf(sqrtf(sp), EPSN);
    e2p[row]     = __expf(INV_T * dp * inv_e * inv_p);
    row_sum[row] = 0.0f;
    pos_sum[row] = 0.0f;
  }
}

// Direct global -> LDS async copy (16B per lane), ASYNCcnt-tracked.
// LDS aperture maps addr[31:0] to the LDS offset, so the low 32 bits of a
// generic pointer to __shared__ are the hardware dsaddr.
__device__ __forceinline__ void async_copy16(void* lds_dst, const void* gsrc) {
  const unsigned lds = (unsigned)(uintptr_t)lds_dst;
  asm volatile("global_load_async_to_lds_b128 %0, %1, off"
               :: "v"(lds), "v"(gsrc) : "memory");
}

// ---------------------------------------------------------------------------
// Kernel B: fused S = exp(2 * en en^T) with diag removal + label masking +
// per-row reduction, via V_WMMA_F32_16X16X4_F32.
// Block: 8 waves; block tile = 128 rows x (8 chunks * 64 cols).
// Wave w owns rows [i0+16w, i0+16w+16): four 16x16 accumulators per chunk.
// A and B tiles staged in LDS (K in halves of 64) via async global->LDS DMA.
// ---------------------------------------------------------------------------
__global__ __launch_bounds__(256) void sim_kernel(
    const float* __restrict__ en, const int* __restrict__ label,
    float* __restrict__ row_sum, float* __restrict__ pos_sum) {
  __shared__ float sA[BLK_ROWS * ASTRIDE];    // 128*68*4 = 34816 B
  __shared__ float sB[CHUNK_COLS * ASTRIDE];  //  64*68*4 = 17408 B

  const int tid  = threadIdx.x;
  const int wave = tid >> 5;
  const int lane = tid & 31;
  const int half = lane >> 4;   // 0: lanes 0-15, 1: lanes 16-31
  const int l16  = lane & 15;

  const int i0     = blockIdx.y * BLK_ROWS;
  const int jpanel = blockIdx.x * (CHUNK_COLS * NCHUNKS);

  // Row labels for this lane's 8 output rows (fixed for the whole block).
  int labr[8];
#pragma unroll
  for (int r = 0; r < 8; ++r)
    labr[r] = label[i0 + wave * 16 + r + 8 * half];

  float rs[8], ps[8];
#pragma unroll
  for (int r = 0; r < 8; ++r) { rs[r] = 0.0f; ps[r] = 0.0f; }

  // WMMA f32 16x16x4 fragment addressing (A and B identical pattern since
  // B[k][n] = en[j0+n][k]): lane = M/N (l16), vgpr pair = K = kk+2*half+{0,1}.
  const float* aptr = &sA[(wave * 16 + l16) * ASTRIDE + 2 * half];
  const float* bptr = &sB[l16 * ASTRIDE + 2 * half];

  for (int c = 0; c < NCHUNKS; ++c) {
    const int j0 = jpanel + c * CHUNK_COLS;

    // Column labels for this chunk (registers, global L0-cached).
    int labc[4];
#pragma unroll
    for (int t = 0; t < 4; ++t) labc[t] = label[j0 + t * 16 + l16];

    v8f acc[4];
#pragma unroll
    for (int t = 0; t < 4; ++t) acc[t] = (v8f){};

    for (int kh = 0; kh < 2; ++kh) {
      __syncthreads();
      // Stage A: 128 rows x 64 K-floats = 2048 x 16B -> 8 async copies/thread.
#pragma unroll
      for (int i = 0; i < 8; ++i) {
        const int flat = tid + i * 256;
        const int row  = flat >> 4;    // 16 float4 per row-half
        const int k4   = flat & 15;
        async_copy16(&sA[row * ASTRIDE + k4 * 4],
                     en + (size_t)(i0 + row) * DDIM + kh * KHALF + k4 * 4);
      }
      // Stage B: 64 rows x 64 K-floats = 1024 x 16B -> 4 async copies/thread.
#pragma unroll
      for (int i = 0; i < 4; ++i) {
        const int flat = tid + i * 256;
        const int row  = flat >> 4;
        const int k4   = flat & 15;
        async_copy16(&sB[row * ASTRIDE + k4 * 4],
                     en + (size_t)(j0 + row) * DDIM + kh * KHALF + k4 * 4);
      }
      asm volatile("s_wait_asynccnt 0" ::: "memory");
      __syncthreads();

#pragma unroll 4
      for (int kk = 0; kk < KHALF; kk += 4) {
        const v2f a = *reinterpret_cast<const v2f*>(aptr + kk);
#pragma unroll
        for (int t = 0; t < 4; ++t) {
          const v2f b = *reinterpret_cast<const v2f*>(bptr + t * 16 * ASTRIDE + kk);
          acc[t] = __builtin_amdgcn_wmma_f32_16x16x4_f32(
              false, a, false, b, (short)0, acc[t], false, false);
        }
      }
    }

    // Branchless epilogue: s = exp(2*dot); mask diagonal + labels.
    // C layout: acc[t][r] = S[i0 + 16w + r + 8*half][j0 + 16t + l16].
#pragma unroll
    for (int t = 0; t < 4; ++t) {
      const int gj = j0 + t * 16 + l16;
#pragma unroll
      for (int r = 0; r < 8; ++r) {
        const int gi   = i0 + wave * 16 + r + 8 * half;
        const float s  = __expf(INV_T * acc[t][r]);
        const bool off = (gi != gj);
        rs[r] += off ? s : 0.0f;
        ps[r] += (off && (labr[r] == labc[t])) ? s : 0.0f;
      }
    }
  }

  // Reduce across the 16 lanes of each half (rows r / r+8), then one atomic
  // per row per array per block.
#pragma unroll
  for (int r = 0; r < 8; ++r) {
    float a = rs[r], b = ps[r];
#pragma unroll
    for (int m = 1; m <= 8; m <<= 1) {
      a += __shfl_xor(a, m, 32);
      b += __shfl_xor(b, m, 32);
    }
    if (l16 == 0) {
      const int gi = i0 + wave * 16 + r + 8 * half;
      atomicAdd(&row_sum[gi], a);
      atomicAdd(&pos_sum[gi], b);
    }
  }
}

// ---------------------------------------------------------------------------
// Kernel C: loss = -mean(log((e2p + pos) / (e2p + tot)))
// ---------------------------------------------------------------------------
__global__ __launch_bounds__(1024) void loss_kernel(
    const float* __restrict__ e2p, const float* __restrict__ row_sum,
    const float* __restrict__ pos_sum, float* __restrict__ out, int n) {
  __shared__ float red[32];
  float acc = 0.0f;
  for (int i = threadIdx.x; i < n; i += 1024) {
    const float ep = e2p[i];
    acc += logf((ep + pos_sum[i]) / (ep + row_sum[i]));
  }
#pragma unroll
  for (int m = 16; m >= 1; m >>= 1) acc += __shfl_xor(acc, m, 32);
  const int lane = threadIdx.x & 31, wave = threadIdx.x >> 5;
  if (lane == 0) red[wave] = acc;
  __syncthreads();
  if (wave == 0) {
    float v = red[lane];
#pragma unroll
    for (int m = 16; m >= 1; m >>= 1) v += __shfl_xor(v, m, 32);
    if (lane == 0) out[0] = -v / (float)n;
  }
}

extern "C" void kernel_launch(void* const* d_in, const int* in_sizes, int n_in,
                              void* d_out, int out_size, void* d_ws, size_t ws_size,
                              hipStream_t stream) {
  (void)in_sizes; (void)n_in; (void)out_size; (void)ws_size;
  const float* embed = (const float*)d_in[0];
  const float* proxy = (const float*)d_in[1];
  const int*   label = (const int*)d_in[2];

  float* en      = (float*)d_ws;                 // 8192*128 floats = 4 MB
  float* e2p     = en + (size_t)NROWS * DDIM;    // 8192
  float* row_sum = e2p + NROWS;                  // 8192
  float* pos_sum = row_sum + NROWS;              // 8192

  normalize_kernel<<<NROWS / 8, 256, 0, stream>>>(embed, proxy, en, e2p,
                                                  row_sum, pos_sum);
  dim3 grid(NROWS / (CHUNK_COLS * NCHUNKS), NROWS / BLK_ROWS);  // (16, 64)
  sim_kernel<<<grid, 256, 0, stream>>>(en, label, row_sum, pos_sum);
  loss_kernel<<<1, 1024, 0, stream>>>(e2p, row_sum, pos_sum, (float*)d_out,
                                      NROWS);
}